// AttnLayer_60636348285537
// MI455X (gfx1250) — compile-verified
//
#include <hip/hip_runtime.h>
#include <hip/hip_bf16.h>

typedef __attribute__((ext_vector_type(2))) float v2f;
typedef __attribute__((ext_vector_type(8))) float v8f;

#define B_    8
#define C_    1024
#define T_    32
#define WH_   196
#define N_    3
#define ROWS  16      // (b,c) rows per block -> one 16-row WMMA M-tile
#define NWAVES 8      // 256 threads / wave32

__global__ __launch_bounds__(256) void attn_gauss_wmma_kernel(
    const float* __restrict__ video,
    const float* __restrict__ mu_t,
    const float* __restrict__ sigma_t,
    float* __restrict__ out)
{
    // A operand: s_tile[m][t], stride 34 (even -> 8B-aligned pairs; 34*l mod 64
    // injective for l in [0,16) -> conflict-free b64 reads).
    __shared__ float s_tile[ROWS][T_ + 2];
    // B operand, pre-padded: gB[kp][2*n+j] = f[n][2*kp+j] (n<3), 0 otherwise.
    __shared__ float gB[T_ / 2][2 * 16];
    __shared__ float f_raw[N_][T_];
    __shared__ float finv[N_];

    const int tid  = (int)threadIdx.x;
    const int lane = tid & 31;
    const int wave = tid >> 5;
    const int r0   = (int)blockIdx.x * ROWS;   // first global (b*C + c) row

    // ---- Phase A: Gaussian attention filters (tiny, per-block) ----
    if (tid < N_ * T_) {
        const int n = tid >> 5;                // T_ == 32
        const int t = tid & 31;
        float mu    = tanhf(mu_t[n]);
        float sg    = 1.0f / (1.0f + expf(-sigma_t[n]));
        float sigma = expf(1.5f - 2.0f * sg);
        float ctr   = (float)(T_ - 1) * (mu + 1.0f) * 0.5f;
        float d     = (float)t - ctr;
        f_raw[n][t] = expf(-(d * d) / (2.0f * sigma * sigma + 1e-16f));
    }
    __syncthreads();
    if (tid < N_) {
        float s = 0.0f;
        #pragma unroll
        for (int t = 0; t < T_; ++t) s += f_raw[tid][t];
        finv[tid] = (1.0f / (s + 1e-16f)) * (1.0f / (float)WH_);
    }
    __syncthreads();
    {   // Build zero-padded, normalized B tile: one float2 per thread.
        const int kp = tid >> 4;      // 0..15 : K-pair index (t = 2*kp, 2*kp+1)
        const int l  = tid & 15;      // 0..15 : B column (n)
        float b0 = 0.0f, b1 = 0.0f;
        if (l < N_) {
            b0 = f_raw[l][2 * kp]     * finv[l];
            b1 = f_raw[l][2 * kp + 1] * finv[l];
        }
        gB[kp][2 * l]     = b0;
        gB[kp][2 * l + 1] = b1;
    }

    // ---- Phase B: spatial (W*H=196) reduction, fully coalesced NT stream ----
    // 512 (row,t) jobs per block; one wave per job; video contiguous per job.
    for (int j = wave; j < ROWS * T_; j += NWAVES) {
        const int row = j >> 5;
        const int t   = j & 31;
        const float* p =
            video + (size_t)((int)blockIdx.x * (ROWS * T_) + j) * WH_;
        float acc = 0.0f;
        for (int k = lane; k < WH_; k += 32)
            acc += __builtin_nontemporal_load(p + k);
        #pragma unroll
        for (int off = 16; off; off >>= 1)
            acc += __shfl_xor(acc, off, 32);
        if (lane == 0) s_tile[row][t] = acc;
    }
    __syncthreads();

    // ---- Phase C: T-contraction on the matrix pipe ----
    // o[m][n] = sum_t s[m][t] * f_hat[n][t] ; 8 chained V_WMMA_F32_16X16X4_F32.
    if (wave == 0) {
        v8f c = {};
        const int half = lane >> 4;   // 0: K pair {0,1}; 1: K pair {2,3}
        const int l    = lane & 15;   // A: row M; B/D: column N
        #pragma unroll
        for (int k = 0; k < T_ / 4; ++k) {
            const int kb = k * 4 + 2 * half;                  // always even
            v2f a = *(const v2f*)&s_tile[l][kb];              // ds_load_b64
            v2f b = *(const v2f*)&gB[kb >> 1][2 * l];         // ds_load_b64
            c = __builtin_amdgcn_wmma_f32_16x16x4_f32(
                    /*neg_a=*/false, a, /*neg_b=*/false, b,
                    /*c_mod=*/(short)0, c, /*reuse_a=*/false, /*reuse_b=*/false);
        }
        // C/D 16x16 f32 layout: VGPR v -> M = v (lanes 0-15) or v+8 (lanes 16-31)
        #pragma unroll
        for (int v = 0; v < 8; ++v) {
            const int m = half * 8 + v;
            if (l < N_) out[(size_t)(r0 + m) * N_ + l] = c[v];
        }
    }
}

extern "C" void kernel_launch(void* const* d_in, const int* in_sizes, int n_in,
                              void* d_out, int out_size, void* d_ws, size_t ws_size,
                              hipStream_t stream) {
    const float* video   = (const float*)d_in[0];
    const float* mu_t    = (const float*)d_in[1];
    const float* sigma_t = (const float*)d_in[2];
    // d_in[3] (meta) unused by the reference forward.
    float* out = (float*)d_out;

    const int blocks = (B_ * C_) / ROWS;   // 512
    attn_gauss_wmma_kernel<<<dim3(blocks), dim3(256), 0, stream>>>(
        video, mu_t, sigma_t, out);
    (void)in_sizes; (void)n_in; (void)out_size; (void)d_ws; (void)ws_size;
}